// All_Graph_Net_5317169512565
// MI455X (gfx1250) — compile-verified
//
#include <hip/hip_runtime.h>

typedef __attribute__((ext_vector_type(2))) float v2f;
typedef __attribute__((ext_vector_type(4))) float v4f;
typedef __attribute__((ext_vector_type(8))) float v8f;

#define FD 128          // feature dim
#define LSTR 132        // padded LDS row stride (floats) -> conflict-free A reads

// ---------------------------------------------------------------------------
// m = relu(h @ Wp^T + bp)          one block = 16 rows x 128 cols, 8 waves
// ---------------------------------------------------------------------------
__global__ __launch_bounds__(256) void gemm_pool_relu(
    const float* __restrict__ h, const float* __restrict__ Wp,
    const float* __restrict__ bp, float* __restrict__ m_out)
{
    __shared__ float tileH[16 * LSTR];
    const int m0 = blockIdx.x * 16;

    // cooperative stage of the 16x128 h tile (8 floats / thread)
    {
        const int t = threadIdx.x;
#pragma unroll
        for (int i = 0; i < 2; ++i) {
            const int idx = t * 8 + i * 4;          // 0..2047
            const int row = idx >> 7, col = idx & 127;
            *(v4f*)&tileH[row * LSTR + col] =
                *(const v4f*)&h[(size_t)(m0 + row) * FD + col];
        }
    }
    __syncthreads();

    const int lane = threadIdx.x & 31;
    const int wave = threadIdx.x >> 5;      // column tile 0..7
    const int n0   = wave * 16;
    const int r    = lane & 15;
    const int kh   = lane >> 4;             // k-half select
    const int kh2  = kh * 2;

    const float* arow = &tileH[r * LSTR + kh2];
    const float* brow = &Wp[(size_t)(n0 + r) * FD + kh2];

    v8f c = {};
#pragma unroll 8
    for (int kb = 0; kb < FD; kb += 4) {
        v2f a = *(const v2f*)(arow + kb);
        v2f b = *(const v2f*)(brow + kb);
        c = __builtin_amdgcn_wmma_f32_16x16x4_f32(false, a, false, b,
                                                  (short)0, c, false, false);
    }

    const float bv = bp[n0 + r];
#pragma unroll
    for (int i = 0; i < 8; ++i) {
        const int row = m0 + i + (kh << 3);
        float v = c[i] + bv;
        m_out[(size_t)row * FD + n0 + r] = v > 0.f ? v : 0.f;
    }
}

// ---------------------------------------------------------------------------
// out = relu(h @ Ws^T + agg @ Wn^T + bias)
// ---------------------------------------------------------------------------
__global__ __launch_bounds__(256) void gemm_out_relu(
    const float* __restrict__ h, const float* __restrict__ agg,
    const float* __restrict__ Ws, const float* __restrict__ Wn,
    const float* __restrict__ bias, float* __restrict__ out)
{
    __shared__ float tileH[16 * LSTR];
    __shared__ float tileA[16 * LSTR];
    const int m0 = blockIdx.x * 16;

    {
        const int t = threadIdx.x;
#pragma unroll
        for (int i = 0; i < 2; ++i) {
            const int idx = t * 8 + i * 4;
            const int row = idx >> 7, col = idx & 127;
            *(v4f*)&tileH[row * LSTR + col] =
                *(const v4f*)&h[(size_t)(m0 + row) * FD + col];
            *(v4f*)&tileA[row * LSTR + col] =
                *(const v4f*)&agg[(size_t)(m0 + row) * FD + col];
        }
    }
    __syncthreads();

    const int lane = threadIdx.x & 31;
    const int wave = threadIdx.x >> 5;
    const int n0   = wave * 16;
    const int r    = lane & 15;
    const int kh   = lane >> 4;
    const int kh2  = kh * 2;

    const float* ah = &tileH[r * LSTR + kh2];
    const float* aa = &tileA[r * LSTR + kh2];
    const float* bs = &Ws[(size_t)(n0 + r) * FD + kh2];
    const float* bn = &Wn[(size_t)(n0 + r) * FD + kh2];

    v8f c = {};
#pragma unroll 4
    for (int kb = 0; kb < FD; kb += 4) {
        v2f a0 = *(const v2f*)(ah + kb);
        v2f b0 = *(const v2f*)(bs + kb);
        c = __builtin_amdgcn_wmma_f32_16x16x4_f32(false, a0, false, b0,
                                                  (short)0, c, false, false);
        v2f a1 = *(const v2f*)(aa + kb);
        v2f b1 = *(const v2f*)(bn + kb);
        c = __builtin_amdgcn_wmma_f32_16x16x4_f32(false, a1, false, b1,
                                                  (short)0, c, false, false);
    }

    const float bv = bias[n0 + r];
#pragma unroll
    for (int i = 0; i < 8; ++i) {
        const int row = m0 + i + (kh << 3);
        float v = c[i] + bv;
        out[(size_t)row * FD + n0 + r] = v > 0.f ? v : 0.f;
    }
}

// ---------------------------------------------------------------------------
// agg init (segment_max identity == 0 since m = relu(...) >= 0)
// ---------------------------------------------------------------------------
__global__ __launch_bounds__(256) void zero_f32(float* __restrict__ p)
{
    const size_t i = ((size_t)blockIdx.x * 256 + threadIdx.x) * 4;
    *(v4f*)(p + i) = (v4f){0.f, 0.f, 0.f, 0.f};
}

// ---------------------------------------------------------------------------
// scatter-max: one wave per edge, 4 floats per lane.
// Values are non-negative (relu output), so uint atomicMax == float max.
// ---------------------------------------------------------------------------
__global__ __launch_bounds__(256) void scatter_max(
    const unsigned int* __restrict__ m, const int* __restrict__ src,
    const int* __restrict__ dst, unsigned int* __restrict__ agg, int E)
{
    const int e = blockIdx.x * 8 + (threadIdx.x >> 5);
    if (e >= E) return;
    const int lane = threadIdx.x & 31;
    const int s = src[e];
    const int d = dst[e];
    const uint4 v = *(const uint4*)(m + (size_t)s * FD + lane * 4);
    unsigned int* o = agg + (size_t)d * FD + lane * 4;
    atomicMax(o + 0, v.x);
    atomicMax(o + 1, v.y);
    atomicMax(o + 2, v.z);
    atomicMax(o + 3, v.w);
}

// ---------------------------------------------------------------------------
extern "C" void kernel_launch(void* const* d_in, const int* in_sizes, int n_in,
                              void* d_out, int out_size, void* d_ws, size_t ws_size,
                              hipStream_t stream)
{
    const float* h_dr  = (const float*)d_in[0];
    const float* h_p   = (const float*)d_in[1];
    const int*   ddi_s = (const int*)d_in[2];
    const int*   ddi_d = (const int*)d_in[3];
    const int*   ppi_s = (const int*)d_in[4];
    const int*   ppi_d = (const int*)d_in[5];
    const float* Wp_d  = (const float*)d_in[6];
    const float* bp_d  = (const float*)d_in[7];
    const float* Ws_d  = (const float*)d_in[8];
    const float* Wn_d  = (const float*)d_in[9];
    const float* b_d   = (const float*)d_in[10];
    const float* Wp_p  = (const float*)d_in[11];
    const float* bp_p  = (const float*)d_in[12];
    const float* Ws_p  = (const float*)d_in[13];
    const float* Wn_p  = (const float*)d_in[14];
    const float* b_p   = (const float*)d_in[15];

    const int nDR  = in_sizes[0] / FD;   // 20000
    const int nP   = in_sizes[1] / FD;   // 40000
    const int eDDI = in_sizes[2];        // 400000
    const int ePPI = in_sizes[4];        // 600000

    float* out   = (float*)d_out;        // [h_dr1 | h_p1 | h_dr2 | h_p2]
    float* h_dr1 = out;
    float* h_p1  = h_dr1 + (size_t)nDR * FD;
    float* h_dr2 = h_p1  + (size_t)nP  * FD;
    float* h_p2  = h_dr2 + (size_t)nDR * FD;

    float* m_buf = (float*)d_ws;                    // <= 40000*128 f32
    float* agg   = m_buf + (size_t)nP * FD;         // <= 40000*128 f32

    auto pool = [&](const float* hin, const int* src, const int* dst,
                    int E, int N, const float* Wp, const float* bp,
                    const float* Ws, const float* Wn, const float* b,
                    float* hout) {
        gemm_pool_relu<<<N / 16, 256, 0, stream>>>(hin, Wp, bp, m_buf);
        zero_f32<<<(N * FD) / 1024, 256, 0, stream>>>(agg);
        scatter_max<<<(E + 7) / 8, 256, 0, stream>>>(
            (const unsigned int*)m_buf, src, dst, (unsigned int*)agg, E);
        gemm_out_relu<<<N / 16, 256, 0, stream>>>(hin, agg, Ws, Wn, b, hout);
    };

    pool(h_dr,  ddi_s, ddi_d, eDDI, nDR, Wp_d, bp_d, Ws_d, Wn_d, b_d, h_dr1);
    pool(h_dr1, ddi_s, ddi_d, eDDI, nDR, Wp_d, bp_d, Ws_d, Wn_d, b_d, h_dr2);
    pool(h_p,   ppi_s, ppi_d, ePPI, nP,  Wp_p, bp_p, Ws_p, Wn_p, b_p, h_p1);
    pool(h_p1,  ppi_s, ppi_d, ePPI, nP,  Wp_p, bp_p, Ws_p, Wn_p, b_p, h_p2);
}